// GINELayer_53197464928897
// MI455X (gfx1250) — compile-verified
//
#include <hip/hip_runtime.h>
#include <hip/hip_bf16.h>

typedef __attribute__((ext_vector_type(2))) float v2f;
typedef __attribute__((ext_vector_type(8))) float v8f;

#define DD      64
#define NNODES  80000
#define NEDGES  1280000
#define BN_EPS_C 1e-5f

// ---------------------------------------------------------------------------
// K0: agg = (1+eps) * x   ; zero stats[0..127]
// ---------------------------------------------------------------------------
__global__ __launch_bounds__(256) void gine_init(const float* __restrict__ x,
                                                 const float* __restrict__ eps,
                                                 float* __restrict__ agg,
                                                 float* __restrict__ stats,
                                                 int total) {
    int i = blockIdx.x * blockDim.x + threadIdx.x;
    if (i < 128) stats[i] = 0.0f;
    if (i < total) {
        agg[i] = (1.0f + eps[0]) * x[i];
    }
}

// ---------------------------------------------------------------------------
// K1: per-edge scatter: agg[dst] += relu(x[src] + emb[attr])
//     16 threads per edge, each does a float4 chunk (4 f32 global atomics)
// ---------------------------------------------------------------------------
__global__ __launch_bounds__(256) void gine_scatter(const float* __restrict__ x,
                                                    const int* __restrict__ edge_index,
                                                    const int* __restrict__ edge_attr,
                                                    const float* __restrict__ emb,
                                                    float* __restrict__ agg,
                                                    int nedges) {
    int idx = blockIdx.x * blockDim.x + threadIdx.x;
    if (idx >= nedges * 16) return;
    int e = idx >> 4;
    int c = (idx & 15) << 2;            // float offset within the 64-d row

    int src  = edge_index[e];
    int dst  = edge_index[nedges + e];
    int attr = edge_attr[e];

    const float4 xv = *(const float4*)(x   + (size_t)src  * DD + c);
    const float4 ev = *(const float4*)(emb + (size_t)attr * DD + c);

    float4 m;
    m.x = fmaxf(xv.x + ev.x, 0.0f);
    m.y = fmaxf(xv.y + ev.y, 0.0f);
    m.z = fmaxf(xv.z + ev.z, 0.0f);
    m.w = fmaxf(xv.w + ev.w, 0.0f);

    float* dp = agg + (size_t)dst * DD + c;
    atomicAdd(dp + 0, m.x);
    atomicAdd(dp + 1, m.y);
    atomicAdd(dp + 2, m.z);
    atomicAdd(dp + 3, m.w);
}

// ---------------------------------------------------------------------------
// WMMA tile conventions (V_WMMA_F32_16X16X4_F32, wave32):
//   lane l: m = l & 15, hh = l >> 4, ka = k + 2*hh (even)
//   A 16x4 : a = {A[m][ka], A[m][ka+1]}           (one b64 global load)
//   B 4x16 : b = {B[ka][n], B[ka+1][n]}, n = m    (one b64 LDS load via
//            paired-K layout: Bs[(k>>1)*128 + n*2 + (k&1)] = B[k][n])
//   C 16x16: acc[v] = C[8*hh + v][n],  v = 0..7
// ---------------------------------------------------------------------------

// K2: h1 = agg @ W1^T + b1 ; accumulate per-feature sum / sumsq into stats
__global__ __launch_bounds__(256) void gine_gemm1(const float* __restrict__ agg,
                                                  const float* __restrict__ W1,
                                                  const float* __restrict__ b1,
                                                  float* __restrict__ h1,
                                                  float* __restrict__ stats) {
    __shared__ __align__(16) float Bs[DD * DD]; // paired-K W1^T, see above
    __shared__ float sb1[DD];
    __shared__ float ssum[DD];
    __shared__ float ssq[DD];

    int tid = threadIdx.x;
    for (int i = tid; i < DD * DD; i += blockDim.x) {
        int j = i >> 6, k = i & 63;              // i = j*64 + k over W1 row-major
        Bs[(k >> 1) * 128 + j * 2 + (k & 1)] = W1[i];   // B[k][j] = W1[j][k]
    }
    if (tid < DD) { sb1[tid] = b1[tid]; ssum[tid] = 0.0f; ssq[tid] = 0.0f; }
    __syncthreads();

    int wave = tid >> 5;
    int lane = tid & 31;
    int m  = lane & 15;
    int hh = lane >> 4;
    int row0 = (blockIdx.x * 8 + wave) * 16;     // 16 rows per wave, exact fit

    v8f acc[4] = {v8f{}, v8f{}, v8f{}, v8f{}};
    const float* arow = agg + (size_t)(row0 + m) * DD;

    for (int k = 0; k < DD; k += 4) {
        int ka = k + 2 * hh;
        v2f a = *(const v2f*)(arow + ka);                     // b64 load
        const v2f* bp = (const v2f*)(Bs + (ka >> 1) * 128);   // paired row
#pragma unroll
        for (int jt = 0; jt < 4; ++jt) {
            v2f b = bp[jt * 16 + m];                          // ds_load_b64
            acc[jt] = __builtin_amdgcn_wmma_f32_16x16x4_f32(
                false, a, false, b, (short)0, acc[jt], false, false);
        }
    }

#pragma unroll
    for (int jt = 0; jt < 4; ++jt) {
        int col = jt * 16 + m;
        float bias = sb1[col];
        float rs = 0.0f, rq = 0.0f;
#pragma unroll
        for (int v = 0; v < 8; ++v) {
            float val = acc[jt][v] + bias;
            h1[(size_t)(row0 + 8 * hh + v) * DD + col] = val;
            rs += val;
            rq += val * val;
        }
        atomicAdd(&ssum[col], rs);   // ds_add_f32
        atomicAdd(&ssq[col], rq);
    }
    __syncthreads();
    if (tid < DD) {
        atomicAdd(&stats[tid], ssum[tid]);
        atomicAdd(&stats[DD + tid], ssq[tid]);
    }
}

// K3: fold raw stats into per-feature affine (scale, shift)
__global__ void gine_bnfinalize(const float* __restrict__ stats,
                                const float* __restrict__ gamma,
                                const float* __restrict__ beta,
                                float* __restrict__ affine) {
    int j = threadIdx.x;
    if (j >= DD) return;
    const float invN = 1.0f / (float)NNODES;
    float mu  = stats[j] * invN;
    float var = stats[DD + j] * invN - mu * mu;
    float sc  = gamma[j] * rsqrtf(var + BN_EPS_C);
    affine[j]      = sc;
    affine[DD + j] = beta[j] - mu * sc;
}

// K4: out = relu(h1*scale + shift) @ W2^T + b2
__global__ __launch_bounds__(256) void gine_gemm2(const float* __restrict__ h1,
                                                  const float* __restrict__ W2,
                                                  const float* __restrict__ b2,
                                                  const float* __restrict__ affine,
                                                  float* __restrict__ out) {
    __shared__ __align__(16) float Bs[DD * DD]; // paired-K W2^T
    __shared__ float sb2[DD];
    __shared__ float ssc[DD];
    __shared__ float ssh[DD];

    int tid = threadIdx.x;
    for (int i = tid; i < DD * DD; i += blockDim.x) {
        int j = i >> 6, k = i & 63;
        Bs[(k >> 1) * 128 + j * 2 + (k & 1)] = W2[i];
    }
    if (tid < DD) { sb2[tid] = b2[tid]; ssc[tid] = affine[tid]; ssh[tid] = affine[DD + tid]; }
    __syncthreads();

    int wave = tid >> 5;
    int lane = tid & 31;
    int m  = lane & 15;
    int hh = lane >> 4;
    int row0 = (blockIdx.x * 8 + wave) * 16;

    v8f acc[4] = {v8f{}, v8f{}, v8f{}, v8f{}};
    const float* arow = h1 + (size_t)(row0 + m) * DD;

    for (int k = 0; k < DD; k += 4) {
        int ka = k + 2 * hh;
        v2f av = *(const v2f*)(arow + ka);                    // b64 load
        v2f a;
        a.x = fmaxf(av.x * ssc[ka]     + ssh[ka],     0.0f);  // BN + ReLU fused
        a.y = fmaxf(av.y * ssc[ka + 1] + ssh[ka + 1], 0.0f);
        const v2f* bp = (const v2f*)(Bs + (ka >> 1) * 128);
#pragma unroll
        for (int jt = 0; jt < 4; ++jt) {
            v2f b = bp[jt * 16 + m];
            acc[jt] = __builtin_amdgcn_wmma_f32_16x16x4_f32(
                false, a, false, b, (short)0, acc[jt], false, false);
        }
    }

#pragma unroll
    for (int jt = 0; jt < 4; ++jt) {
        int col = jt * 16 + m;
        float bias = sb2[col];
#pragma unroll
        for (int v = 0; v < 8; ++v) {
            out[(size_t)(row0 + 8 * hh + v) * DD + col] = acc[jt][v] + bias;
        }
    }
}

// ---------------------------------------------------------------------------
extern "C" void kernel_launch(void* const* d_in, const int* in_sizes, int n_in,
                              void* d_out, int out_size, void* d_ws, size_t ws_size,
                              hipStream_t stream) {
    const float* x        = (const float*)d_in[0];
    const int*   eidx     = (const int*)d_in[1];
    const int*   eattr    = (const int*)d_in[2];
    const float* emb      = (const float*)d_in[3];
    const float* W1       = (const float*)d_in[4];
    const float* b1       = (const float*)d_in[5];
    const float* gamma    = (const float*)d_in[6];
    const float* beta     = (const float*)d_in[7];
    const float* W2       = (const float*)d_in[8];
    const float* b2       = (const float*)d_in[9];
    const float* eps      = (const float*)d_in[10];

    float* ws     = (float*)d_ws;
    float* agg    = ws;                                  // N*D
    float* h1     = agg + (size_t)NNODES * DD;           // N*D
    float* stats  = h1 + (size_t)NNODES * DD;            // 128 (sum, sumsq)
    float* affine = stats + 128;                         // 128 (scale, shift)
    float* outp   = (float*)d_out;

    const int total = NNODES * DD;                       // 5,120,000
    gine_init<<<(total + 255) / 256, 256, 0, stream>>>(x, eps, agg, stats, total);

    const int sthreads = NEDGES * 16;                    // 20,480,000
    gine_scatter<<<(sthreads + 255) / 256, 256, 0, stream>>>(x, eidx, eattr, emb, agg, NEDGES);

    const int gblocks = NNODES / 128;                    // 625, exact
    gine_gemm1<<<gblocks, 256, 0, stream>>>(agg, W1, b1, h1, stats);

    gine_bnfinalize<<<1, 64, 0, stream>>>(stats, gamma, beta, affine);

    gine_gemm2<<<gblocks, 256, 0, stream>>>(h1, W2, b2, affine, outp);
}